// EIGNLayer_78700980732026
// MI455X (gfx1250) — compile-verified
//
#include <hip/hip_runtime.h>
#include <math.h>

// EIGN layer for MI455X (gfx1250, wave32).
// Pipeline per phase: zero accumulators -> COO SpMM scatter (f32 HW atomics,
// one wave per nonzero) -> fused triple-GEMM (V_WMMA_F32_16X16X4_F32) +
// LayerNorm + exact GELU.
// Workspace layout: [B0: N*128 f32][B1: N*128 f32][6x transposed 128x128 W].

#define DD 128
#define LSTR 132   // LDS row stride (floats) to avoid 64-bank conflicts

typedef __attribute__((ext_vector_type(2))) float v2f;
typedef __attribute__((ext_vector_type(8))) float v8f;

__global__ __launch_bounds__(256) void zero_f32(float* __restrict__ p, long long n) {
  long long i = (long long)blockIdx.x * blockDim.x + threadIdx.x;
  long long stride = (long long)gridDim.x * blockDim.x;
  for (; i < n; i += stride) p[i] = 0.0f;
}

// Wt[k][j] = W[j][k], 128x128
__global__ __launch_bounds__(256) void transpose_w(const float* __restrict__ W,
                                                   float* __restrict__ Wt) {
  int t = blockIdx.x * blockDim.x + threadIdx.x;   // 16384 threads
  int j = t >> 7, k = t & 127;
  Wt[k * DD + j] = W[j * DD + k];
}

// One wave32 per nonzero: 32 lanes x float4 = one full 128-float row.
// out[r,:] += v * X[c,:]   via hardware global_atomic_add_f32 (L2-resident).
__global__ __launch_bounds__(256) void spmm_scatter(
    const int* __restrict__ rows, const int* __restrict__ cols,
    const float* __restrict__ vals, const float* __restrict__ X,
    float* __restrict__ out, int nnz) {
  long long t = (long long)blockIdx.x * blockDim.x + threadIdx.x;
  int e = (int)(t >> 5);
  int q = (int)(t & 31);
  if (e >= nnz) return;
  int r = rows[e];
  int c = cols[e];
  float v = vals[e];
  const float4 x = *(const float4*)(X + (long long)c * DD + q * 4);
  float* o = out + (long long)r * DD + q * 4;
  unsafeAtomicAdd(o + 0, v * x.x);
  unsafeAtomicAdd(o + 1, v * x.y);
  unsafeAtomicAdd(o + 2, v * x.z);
  unsafeAtomicAdd(o + 3, v * x.w);
}

// Per block: 16 rows. 8 waves, wave w owns columns [16w, 16w+16).
// acc = S0 @ W0.T + S1 @ W1.T + X @ W2.T  via chained f32 WMMA,
// then row-wise LayerNorm (mean/var over 128 cols) + exact GELU.
// SK/SN are COMPILE-TIME so B loads become a single per-lane base address
// + immediate-offset global_load_b32 (max offset 127*512B < 24-bit imm).
//   SK=128, SN=1   : transposed weights (Wt[k][j])
//   SK=1,   SN=128 : raw nn.Linear weights (W[j][k]) fallback
template <int SK, int SN>
__global__ __launch_bounds__(256, 2) void gemm3_ln_gelu(
    const float* __restrict__ S0, const float* __restrict__ S1,
    const float* __restrict__ X,
    const float* __restrict__ W0, const float* __restrict__ W1,
    const float* __restrict__ W2,
    const float* __restrict__ gamma, const float* __restrict__ beta,
    float* __restrict__ out, int n_rows) {
  __shared__ float sA[3][16][LSTR];   // staged activation tiles (16x128)
  __shared__ float sO[16][LSTR];      // pre-LN output tile (16x128)
  __shared__ float red0[16][17];
  __shared__ float red1[16][17];
  __shared__ float mu_s[16], rs_s[16];

  const int tid = threadIdx.x;
  const int wave = tid >> 5;
  const int lane = tid & 31;
  const int row0 = blockIdx.x * 16;

  // ---- Stage S0, S1, X tiles into LDS (coalesced float4) ----
  const float* srcs[3] = {S0, S1, X};
#pragma unroll
  for (int m = 0; m < 3; ++m) {
    const float* src = srcs[m];
#pragma unroll
    for (int it = 0; it < 2; ++it) {
      int idx = it * 256 + tid;       // 512 float4 per 16x128 tile
      int r = idx >> 5;
      int c4 = idx & 31;
      int grow = row0 + r;
      float4 val = make_float4(0.f, 0.f, 0.f, 0.f);
      if (grow < n_rows) val = *(const float4*)(src + (long long)grow * DD + c4 * 4);
      float* dst = &sA[m][r][c4 * 4];
      dst[0] = val.x; dst[1] = val.y; dst[2] = val.z; dst[3] = val.w;
    }
  }
  __syncthreads();

  // ---- Triple GEMM via V_WMMA_F32_16X16X4_F32 ----
  // A 16x4 layout: lanes 0-15 -> M=lane (K = k+0,k+1); lanes 16-31 -> K = k+2,k+3.
  // B 4x16 layout mirrors A; C/D: 8 VGPRs, lanes 16-31 hold M=8..15.
  const int col0 = wave * 16;
  const int lidx = lane & 15;
  const int khalf = (lane >> 4) << 1;   // 0 or 2

  v8f acc = {0.f, 0.f, 0.f, 0.f, 0.f, 0.f, 0.f, 0.f};
  const float* wts[3] = {W0, W1, W2};
#pragma unroll
  for (int m = 0; m < 3; ++m) {
    // Per-lane base: element B[khalf][col0+lidx]; k-step advances by 4*SK,
    // b.y is one K-row further (+SK). All offsets compile-time immediates.
    const float* __restrict__ Wp = wts[m] + khalf * SK + (col0 + lidx) * SN;
    const float* __restrict__ Ap = &sA[m][lidx][khalf];
#pragma unroll 8
    for (int k = 0; k < DD; k += 4) {
      v2f a, b;
      a.x = Ap[k + 0];
      a.y = Ap[k + 1];
      b.x = Wp[k * SK];
      b.y = Wp[k * SK + SK];
      acc = __builtin_amdgcn_wmma_f32_16x16x4_f32(
          /*neg_a=*/false, a, /*neg_b=*/false, b,
          /*c_mod=*/(short)0, acc, /*reuse_a=*/false, /*reuse_b=*/false);
    }
  }

  // ---- Spill C/D tile to LDS for the row-wise LayerNorm ----
  {
    int base_r = (lane >> 4) * 8;     // lanes 16-31 hold M=8..15
#pragma unroll
    for (int j = 0; j < 8; ++j) sO[base_r + j][col0 + lidx] = acc[j];
  }
  __syncthreads();

  // ---- LayerNorm reduction: 16 threads per row, 8 cols each ----
  {
    int r = tid >> 4, s = tid & 15;
    float ps = 0.f, ps2 = 0.f;
#pragma unroll
    for (int j = 0; j < 8; ++j) {
      float x = sO[r][s * 8 + j];
      ps += x;
      ps2 += x * x;
    }
    red0[r][s] = ps;
    red1[r][s] = ps2;
  }
  __syncthreads();
  if (tid < 16) {
    float s1 = 0.f, s2 = 0.f;
#pragma unroll
    for (int s = 0; s < 16; ++s) { s1 += red0[tid][s]; s2 += red1[tid][s]; }
    float mu = s1 * (1.0f / 128.0f);
    float var = s2 * (1.0f / 128.0f) - mu * mu;
    mu_s[tid] = mu;
    rs_s[tid] = rsqrtf(var + 1e-5f);
  }
  __syncthreads();

  // ---- Normalize + exact GELU + coalesced store ----
#pragma unroll
  for (int it = 0; it < 8; ++it) {
    int idx = it * 256 + tid;         // 2048 elems / 256 threads
    int r = idx >> 7;
    int c = idx & 127;
    int grow = row0 + r;
    if (grow < n_rows) {
      float x = sO[r][c];
      float y = (x - mu_s[r]) * rs_s[r] * gamma[c] + beta[c];
      float g = 0.5f * y * (1.0f + erff(y * 0.70710678118654752f));
      out[(long long)grow * DD + c] = g;
    }
  }
}

extern "C" void kernel_launch(void* const* d_in, const int* in_sizes, int n_in,
                              void* d_out, int out_size, void* d_ws, size_t ws_size,
                              hipStream_t stream) {
  const float* X_equ = (const float*)d_in[0];
  const float* X_inv = (const float*)d_in[1];
  const int*   le_r  = (const int*)d_in[2];
  const int*   le_c  = (const int*)d_in[3];
  const float* le_v  = (const float*)d_in[4];
  const int*   ie_r  = (const int*)d_in[5];
  const int*   ie_c  = (const int*)d_in[6];
  const float* ie_v  = (const float*)d_in[7];
  const int*   li_r  = (const int*)d_in[8];
  const int*   li_c  = (const int*)d_in[9];
  const float* li_v  = (const float*)d_in[10];
  const int*   ei_r  = (const int*)d_in[11];
  const int*   ei_c  = (const int*)d_in[12];
  const float* ei_v  = (const float*)d_in[13];
  const float* W1 = (const float*)d_in[14];
  const float* W2 = (const float*)d_in[15];
  const float* W3 = (const float*)d_in[16];
  const float* W4 = (const float*)d_in[17];
  const float* W5 = (const float*)d_in[18];
  const float* W6 = (const float*)d_in[19];
  const float* g_equ = (const float*)d_in[20];
  const float* b_equ = (const float*)d_in[21];
  const float* g_inv = (const float*)d_in[22];
  const float* b_inv = (const float*)d_in[23];

  const int N = in_sizes[0] / DD;      // 100000
  const int nnz_le = in_sizes[2];
  const int nnz_ie = in_sizes[5];
  const int nnz_li = in_sizes[8];
  const int nnz_ei = in_sizes[11];

  float* B0 = (float*)d_ws;
  float* B1 = B0 + (size_t)N * DD;
  const size_t acc_bytes = 2ull * (size_t)N * DD * sizeof(float);

  // Transpose weights once into workspace if it fits (immediate-offset
  // coalesced B reads), else raw-layout fallback. Deterministic either way.
  const float* Wsrc[6] = {W1, W2, W5, W3, W4, W6};
  const float* Wuse[6];
  const bool transposed =
      (ws_size >= acc_bytes + 6ull * DD * DD * sizeof(float));
  if (transposed) {
    float* wt = (float*)((char*)d_ws + acc_bytes);
    for (int i = 0; i < 6; ++i) {
      float* dst = wt + (size_t)i * DD * DD;
      transpose_w<<<(DD * DD) / 256, 256, 0, stream>>>(Wsrc[i], dst);
      Wuse[i] = dst;
    }
  } else {
    for (int i = 0; i < 6; ++i) Wuse[i] = Wsrc[i];
  }

  const long long accN = 2ll * N * DD;
  float* out_equ = (float*)d_out;
  float* out_inv = out_equ + (size_t)N * DD;
  const int gemm_blocks = (N + 15) / 16;

  // ---------------- Phase E (equivariant stream) ----------------
  zero_f32<<<2048, 256, 0, stream>>>(B0, accN);   // zeros B0 and B1
  {
    long long t0 = (long long)nnz_le * 32;
    spmm_scatter<<<(int)((t0 + 255) / 256), 256, 0, stream>>>(le_r, le_c, le_v, X_equ, B0, nnz_le);
    long long t1 = (long long)nnz_ie * 32;
    spmm_scatter<<<(int)((t1 + 255) / 256), 256, 0, stream>>>(ie_r, ie_c, ie_v, X_inv, B1, nnz_ie);
  }
  if (transposed) {
    gemm3_ln_gelu<DD, 1><<<gemm_blocks, 256, 0, stream>>>(
        B0, B1, X_equ, Wuse[0], Wuse[1], Wuse[2], g_equ, b_equ, out_equ, N);
  } else {
    gemm3_ln_gelu<1, DD><<<gemm_blocks, 256, 0, stream>>>(
        B0, B1, X_equ, Wuse[0], Wuse[1], Wuse[2], g_equ, b_equ, out_equ, N);
  }

  // ---------------- Phase I (invariant stream) ----------------
  zero_f32<<<2048, 256, 0, stream>>>(B0, accN);
  {
    long long t0 = (long long)nnz_li * 32;
    spmm_scatter<<<(int)((t0 + 255) / 256), 256, 0, stream>>>(li_r, li_c, li_v, X_inv, B0, nnz_li);
    long long t1 = (long long)nnz_ei * 32;
    spmm_scatter<<<(int)((t1 + 255) / 256), 256, 0, stream>>>(ei_r, ei_c, ei_v, X_equ, B1, nnz_ei);
  }
  if (transposed) {
    gemm3_ln_gelu<DD, 1><<<gemm_blocks, 256, 0, stream>>>(
        B0, B1, X_inv, Wuse[3], Wuse[4], Wuse[5], g_inv, b_inv, out_inv, N);
  } else {
    gemm3_ln_gelu<1, DD><<<gemm_blocks, 256, 0, stream>>>(
        B0, B1, X_inv, Wuse[3], Wuse[4], Wuse[5], g_inv, b_inv, out_inv, N);
  }
}